// ROIAlignBatch_69475390980647
// MI455X (gfx1250) — compile-verified
//
#include <hip/hip_runtime.h>
#include <stdint.h>

// Problem constants (from reference setup_inputs)
#define BATCH 8
#define CHANS 256
#define FH    100
#define FW    152
#define KROI  128
#define POOLH 7
#define POOLW 7
#define HWPIX (FH * FW)           // 15200 = 475 * 32 exactly
#define NBIN  (POOLH * POOLW)     // 49
#define GH    (POOLH * 2)         // 14
#define GW    (POOLW * 2)         // 14
#define NSAMP (GH * GW)           // 196

typedef float v4f __attribute__((ext_vector_type(4)));   // native vector for NT stores

// ---------------------------------------------------------------------------
// CDNA5 async global->LDS helpers (gfx1250). ASYNCcnt tracked; waited with
// s_wait_asynccnt. the low 32 bits of the generic shared ptr are the LDS destination address.
// NT hint: NCHW source is streamed exactly once -> keep L2 for the NHWC copy.
// ---------------------------------------------------------------------------
__device__ __forceinline__ void async_g2lds_b128_nt(void* lds_dst, const void* gsrc) {
    uint32_t la = (uint32_t)(uintptr_t)lds_dst;
    asm volatile("global_load_async_to_lds_b128 %0, %1, off th:TH_LOAD_NT"
                 :: "v"(la), "v"(gsrc) : "memory");
}
__device__ __forceinline__ void wait_async0() {
    asm volatile("s_wait_asynccnt 0x0" ::: "memory");
}

// ---------------------------------------------------------------------------
// Kernel 1: NCHW -> NHWC transpose. Tile: 32 channels x 32 pixels.
// Each thread issues exactly ONE async b128 (global -> LDS direct), so a wave
// moves 512 B per instruction. Store phase is coalesced along C.
// grid = (HWPIX/32, CHANS/32, BATCH), block = 256 (8 waves).
// LDS row stride 36 floats = 144 B: keeps 16 B alignment for b128 async writes,
// transposed reads see only 2-way bank conflicts.
// ---------------------------------------------------------------------------
__global__ __launch_bounds__(256)
void nchw_to_nhwc_kernel(const float* __restrict__ in, float* __restrict__ out) {
    __shared__ float tile[32 * 36];

    const int t  = threadIdx.x;
    const int ch = t >> 3;                   // 0..31  channel within tile
    const int ck = t & 7;                    // 0..7   4-pixel chunk
    const int pt = blockIdx.x;               // pixel tile (0..474)
    const int ct = blockIdx.y;               // channel tile (0..7)
    const int b  = blockIdx.z;

    const float* src = in + ((size_t)(b * CHANS + ct * 32 + ch)) * HWPIX
                          + pt * 32 + ck * 4;
    // copy 4 pixels of input channel ct*32+ch from global memory into tile row ch, cols ck*4..ck*4+3
    async_g2lds_b128_nt(&tile[ch * 36 + ck * 4], src);
    wait_async0();
    __syncthreads();

    const int tx = t & 31;                   // channel lane for store
    const int ty = t >> 5;                   // 0..7
    float* dst = out + ((size_t)b * HWPIX + (size_t)pt * 32) * CHANS
                     + (size_t)ct * 32 + tx;
#pragma unroll
    for (int i = 0; i < 4; ++i) {
        const int pr = ty + i * 8;           // pixel row within tile
        dst[(size_t)pr * CHANS] = tile[tx * 36 + pr];
    }
}

// ---------------------------------------------------------------------------
// Kernel 2: ROI-Align gather. One block (64 threads = 2 waves) per ROI.
// Thread owns 4 consecutive channels -> every tap is a float4 (b128) load,
// 512 B per wave instruction. Bilinear offsets/weights are computed once per
// ROI into LDS (broadcast reads in the main loop). Results staged in LDS so
// the final store is a contiguous non-temporal burst matching out layout
// (c*49 + bin per ROI).
// ---------------------------------------------------------------------------
template <bool NHWC>
__global__ __launch_bounds__(64)
void roi_align_kernel(const float* __restrict__ feat,
                      const float* __restrict__ rois,
                      const int*   __restrict__ stride_p,
                      float* __restrict__ out) {
    __shared__ float sb[NBIN * CHANS];       // 50176 B, layout [c][bin]
    __shared__ int   soff[NSAMP * 4];        // 4 tap offsets per sample
    __shared__ float swgt[NSAMP * 4];        // 4 tap weights per sample

    const int roi = blockIdx.x;              // 0..1023
    const int b   = roi / KROI;
    const int tid = threadIdx.x;             // 0..63
    const int c0  = tid * 4;                 // first owned channel

    const float scale = 1.0f / (float)stride_p[0];
    const float x1 = rois[roi * 4 + 0];
    const float y1 = rois[roi * 4 + 1];
    const float x2 = rois[roi * 4 + 2];
    const float y2 = rois[roi * 4 + 3];

    const float start_w = x1 * scale - 0.5f;
    const float start_h = y1 * scale - 0.5f;
    const float bin_w   = (x2 - x1) * scale / (float)POOLW;
    const float bin_h   = (y2 - y1) * scale / (float)POOLH;

    // --- Precompute per-sample tap offsets + bilinear weights (once/ROI) ---
    for (int s = tid; s < NSAMP; s += 64) {
        const int gh = s / GW;
        const int gw = s - gh * GW;
        const float yy = start_h + ((float)gh + 0.5f) * 0.5f * bin_h;
        const float xx = start_w + ((float)gw + 0.5f) * 0.5f * bin_w;
        const bool valid = (yy >= -1.0f) && (yy <= (float)FH) &&
                           (xx >= -1.0f) && (xx <= (float)FW);
        const float yc = fminf(fmaxf(yy, 0.0f), (float)(FH - 1));
        const float xc = fminf(fmaxf(xx, 0.0f), (float)(FW - 1));
        const int y0 = (int)floorf(yc);
        const int x0 = (int)floorf(xc);
        const float ly = yc - (float)y0;
        const float lx = xc - (float)x0;
        const int y1i = min(y0 + 1, FH - 1);
        const int x1i = min(x0 + 1, FW - 1);
        const float hy = 1.0f - ly, hx = 1.0f - lx;
        float w00 = hy * hx, w01 = hy * lx, w10 = ly * hx, w11 = ly * lx;
        if (!valid) { w00 = w01 = w10 = w11 = 0.0f; }   // branch-free taps

        const int mult = NHWC ? CHANS : 1;   // element offset within batch plane
        soff[s * 4 + 0] = (y0  * FW + x0 ) * mult;
        soff[s * 4 + 1] = (y0  * FW + x1i) * mult;
        soff[s * 4 + 2] = (y1i * FW + x0 ) * mult;
        soff[s * 4 + 3] = (y1i * FW + x1i) * mult;
        swgt[s * 4 + 0] = w00;
        swgt[s * 4 + 1] = w01;
        swgt[s * 4 + 2] = w10;
        swgt[s * 4 + 3] = w11;
    }
    __syncthreads();

    const float* fb = NHWC ? feat + (size_t)b * HWPIX * CHANS + c0 : feat;

    for (int ph = 0; ph < POOLH; ++ph) {
        for (int pw = 0; pw < POOLW; ++pw) {
            v4f acc = (v4f)(0.0f);
#pragma unroll
            for (int sy = 0; sy < 2; ++sy) {
#pragma unroll
                for (int sx = 0; sx < 2; ++sx) {
                    const int s = (ph * 2 + sy) * GW + (pw * 2 + sx);
                    const int4 off = *(const int4*)&soff[s * 4];     // LDS b128 broadcast
                    const v4f  w   = *(const v4f*) &swgt[s * 4];     // LDS b128 broadcast
                    if (NHWC) {
                        const v4f v0 = *(const v4f*)(fb + off.x);
                        const v4f v1 = *(const v4f*)(fb + off.y);
                        const v4f v2 = *(const v4f*)(fb + off.z);
                        const v4f v3 = *(const v4f*)(fb + off.w);
                        acc += w.x * v0 + w.y * v1 + w.z * v2 + w.w * v3;
                    } else {
#pragma unroll
                        for (int j = 0; j < 4; ++j) {
                            const float* pc = feat + (size_t)(b * CHANS + c0 + j) * HWPIX;
                            acc[j] += w.x * pc[off.x] + w.y * pc[off.y]
                                    + w.z * pc[off.z] + w.w * pc[off.w];
                        }
                    }
                }
            }
            const int bin = ph * POOLW + pw;
            sb[(c0 + 0) * NBIN + bin] = acc.x * 0.25f;
            sb[(c0 + 1) * NBIN + bin] = acc.y * 0.25f;
            sb[(c0 + 2) * NBIN + bin] = acc.z * 0.25f;
            sb[(c0 + 3) * NBIN + bin] = acc.w * 0.25f;
        }
    }
    __syncthreads();

    // Contiguous non-temporal burst: 12544 floats (= 3136 v4f) per ROI.
    v4f* ob = (v4f*)(out + (size_t)roi * CHANS * NBIN);
    const v4f* sv = (const v4f*)sb;
    for (int i = tid; i < (NBIN * CHANS) / 4; i += 64) {
        __builtin_nontemporal_store(sv[i], ob + i);
    }
}

// ---------------------------------------------------------------------------
// Launcher
// ---------------------------------------------------------------------------
extern "C" void kernel_launch(void* const* d_in, const int* in_sizes, int n_in,
                              void* d_out, int out_size, void* d_ws, size_t ws_size,
                              hipStream_t stream) {
    const float* feat    = (const float*)d_in[0];
    const float* rois    = (const float*)d_in[1];
    const int*   stridep = (const int*)d_in[2];
    float* out = (float*)d_out;

    const size_t nhwc_bytes = (size_t)BATCH * HWPIX * CHANS * sizeof(float);

    if (ws_size >= nhwc_bytes) {
        float* nhwc = (float*)d_ws;
        dim3 tg(HWPIX / 32, CHANS / 32, BATCH);   // (475, 8, 8)
        hipLaunchKernelGGL(nchw_to_nhwc_kernel, tg, dim3(256), 0, stream, feat, nhwc);
        hipLaunchKernelGGL((roi_align_kernel<true>), dim3(BATCH * KROI), dim3(64),
                           0, stream, nhwc, rois, stridep, out);
    } else {
        hipLaunchKernelGGL((roi_align_kernel<false>), dim3(BATCH * KROI), dim3(64),
                           0, stream, feat, rois, stridep, out);
    }
}